// S5FR_78907139162801
// MI455X (gfx1250) — compile-verified
//
#include <hip/hip_runtime.h>

typedef __attribute__((ext_vector_type(2))) float v2f;
typedef __attribute__((ext_vector_type(8))) float v8f;

#define Hh 128
#define Pp 128
#define Bb 8
#define Ll 4096
#define LF 2049
#define NCOLS (Bb * LF) /* 16392 */
#define PI_F 3.14159265358979323846f

// ---------------- deinterleave (n,2) float -> planar ----------------
__global__ void deinterleave_kernel(const float* __restrict__ src,
                                    float* __restrict__ dr,
                                    float* __restrict__ di, int n) {
  int i = blockIdx.x * blockDim.x + threadIdx.x;
  if (i < n) { dr[i] = src[2 * i]; di[i] = src[2 * i + 1]; }
}

// ---------------- 128-point DFT matrices: W = exp(-2pi i hk/128), WI = conj(W)/128 ----------------
__global__ void dft_build_kernel(float* __restrict__ Wr, float* __restrict__ Wi,
                                 float* __restrict__ WIr, float* __restrict__ WIi) {
  int i = blockIdx.x * blockDim.x + threadIdx.x;
  if (i >= Hh * Hh) return;
  int h = i >> 7, k = i & 127;
  int t = (h * k) & 127;  // hk mod 128 (twiddle periodicity)
  float ang = -2.0f * PI_F * (float)t / 128.0f;
  float s, c;
  __sincosf(ang, &s, &c);
  Wr[i] = c;  Wi[i] = s;
  WIr[i] = c * (1.0f / 128.0f);
  WIi[i] = -s * (1.0f / 128.0f);
}

// ---------------- complex GEMM via V_WMMA_F32_16X16X4_F32 ----------------
// Z (128 x N) = A (128x128) * X (128 x N)  [+ Z if accumulate], planar complex.
// One wave per 16x16 output tile. 4 real WMMA chains: RR, II, RI, IR.
// Out-of-range columns are CLAMPED (not zero-filled): a B-fragment lane's data
// only affects output column n, which is masked at the store, so clamping keeps
// the k-loop branch-free and EXEC all-ones throughout.
__global__ void cgemm_wmma_kernel(const float* __restrict__ Ar, const float* __restrict__ Ai,
                                  const float* __restrict__ Xr, const float* __restrict__ Xi,
                                  float* __restrict__ Zr, float* __restrict__ Zi,
                                  int N, int accumulate) {
  const int lane = threadIdx.x & 31;
  const int rowTile = blockIdx.x << 4;
  const int colTile = blockIdx.y << 4;
  // A fragment (16x4 f32): lane holds row m = lane%16, K pair koff = 2*(lane/16)
  const int m = rowTile + (lane & 15);
  const int koff = (lane >> 4) << 1;
  // B fragment (4x16): lane holds col n = lane%16, same K pair
  const int n = colTile + (lane & 15);
  const int nc = (n < N) ? n : (N - 1);  // clamp: garbage column never stored

  const float* pAr = Ar + m * 128 + koff;
  const float* pAi = Ai + m * 128 + koff;
  const float* pXr = Xr + (size_t)koff * N + nc;
  const float* pXi = Xi + (size_t)koff * N + nc;
  const size_t xstep = (size_t)4 * N;

  v8f accRR = {}, accII = {}, accRI = {}, accIR = {};
#pragma unroll 4
  for (int k = 0; k < 32; ++k) {
    v2f a_r, a_i, x_r, x_i;
    a_r[0] = pAr[0]; a_r[1] = pAr[1];
    a_i[0] = pAi[0]; a_i[1] = pAi[1];
    x_r[0] = pXr[0]; x_r[1] = pXr[N];
    x_i[0] = pXi[0]; x_i[1] = pXi[N];
    accRR = __builtin_amdgcn_wmma_f32_16x16x4_f32(false, a_r, false, x_r, (short)0, accRR, false, false);
    accII = __builtin_amdgcn_wmma_f32_16x16x4_f32(false, a_i, false, x_i, (short)0, accII, false, false);
    accRI = __builtin_amdgcn_wmma_f32_16x16x4_f32(false, a_r, false, x_i, (short)0, accRI, false, false);
    accIR = __builtin_amdgcn_wmma_f32_16x16x4_f32(false, a_i, false, x_r, (short)0, accIR, false, false);
    pAr += 4; pAi += 4;
    pXr += xstep; pXi += xstep;
  }
  // C/D layout: VGPR v -> row = rowTile + v + 8*(lane/16), col = colTile + lane%16
  const int outRow = rowTile + ((lane >> 4) << 3);
  const int outCol = colTile + (lane & 15);
  if (outCol < N) {
    size_t idx = (size_t)outRow * N + outCol;
    for (int v = 0; v < 8; ++v, idx += N) {
      float zr = accRR[v] - accII[v];
      float zi = accRI[v] + accIR[v];
      if (accumulate) { zr += Zr[idx]; zi += Zi[idx]; }
      Zr[idx] = zr;
      Zi[idx] = zi;
    }
  }
}

// ---------------- in-LDS radix-2 Stockham FFT, length 4096, 512 threads ----------------
__device__ __forceinline__ void fft4096_lds(float* sr0, float* si0,
                                            float* sr1, float* si1,
                                            float sign, int tid) {
  float *srA = sr0, *siA = si0, *srB = sr1, *siB = si1;
  for (int s = 0; s < 12; ++s) {
    const int Ls = 1 << s;
    for (int j = tid; j < 2048; j += 512) {
      int offs = j & (Ls - 1);
      int base2 = (j - offs) << 1;
      float ar = srA[j],        ai = siA[j];
      float br = srA[j + 2048], bi = siA[j + 2048];
      float ang = sign * PI_F * (float)offs / (float)Ls;
      float sn, cs;
      __sincosf(ang, &sn, &cs);
      float tr = cs * br - sn * bi;
      float ti = cs * bi + sn * br;
      srB[base2 + offs]      = ar + tr;
      siB[base2 + offs]      = ai + ti;
      srB[base2 + offs + Ls] = ar - tr;
      siB[base2 + offs + Ls] = ai - ti;
    }
    __syncthreads();
    float* t;
    t = srA; srA = srB; srB = t;
    t = siA; siA = siB; siB = t;
  }
  // 12 stages (even) -> result back in sr0/si0
}

// ---------------- forward: u(b,h,:) -> V rows (h), cols (b*LF + l), planar ----------------
__global__ void __launch_bounds__(512) rfft_fwd_kernel(const float* __restrict__ u,
                                                       float* __restrict__ Vr,
                                                       float* __restrict__ Vi) {
  __shared__ float lds[4 * 4096];
  float* sr0 = lds;
  float* si0 = lds + 4096;
  float* sr1 = lds + 8192;
  float* si1 = lds + 12288;
  const int tid = threadIdx.x;
  const int b = blockIdx.x >> 7;
  const int h = blockIdx.x & 127;
  const float* up = u + ((size_t)(b * Hh + h)) * Ll;
  for (int i = tid; i < 4096; i += 512) { sr0[i] = up[i]; si0[i] = 0.0f; }
  __syncthreads();
  fft4096_lds(sr0, si0, sr1, si1, -1.0f, tid);
  const size_t rowbase = (size_t)h * NCOLS + (size_t)b * LF;
  for (int l = tid; l < LF; l += 512) {
    Vr[rowbase + l] = sr0[l];
    Vi[rowbase + l] = si0[l];
  }
}

// ---------------- Cauchy kernel multiply (in place on T): T *= 1/(i*w_l - Lambda_bar_p) ----------------
__global__ void mid_mul_kernel(float* __restrict__ Tr, float* __restrict__ Ti,
                               const float* __restrict__ Lam_ri) {
  size_t i = (size_t)blockIdx.x * blockDim.x + threadIdx.x;
  if (i >= (size_t)Pp * NCOLS) return;
  int p = (int)(i / NCOLS);
  int c = (int)(i % NCOLS);
  int l = c % LF;
  float w = 2.0f * PI_F * (float)l / (float)Ll;
  float lr = Lam_ri[2 * p], li = Lam_ri[2 * p + 1];
  float dr = -lr;
  float di = w - li;
  float inv = 1.0f / (dr * dr + di * di);
  float mr = dr * inv;
  float mi = -di * inv;
  float tr = Tr[i], ti = Ti[i];
  Tr[i] = mr * tr - mi * ti;
  Ti[i] = mr * ti + mi * tr;
}

// ---------------- inverse: Hermitian reconstruct, IFFT, 1/L, exact GELU ----------------
__global__ void __launch_bounds__(512) irfft_gelu_kernel(const float* __restrict__ Yr,
                                                         const float* __restrict__ Yi,
                                                         float* __restrict__ out) {
  __shared__ float lds[4 * 4096];
  float* sr0 = lds;
  float* si0 = lds + 4096;
  float* sr1 = lds + 8192;
  float* si1 = lds + 12288;
  const int tid = threadIdx.x;
  const int b = blockIdx.x >> 7;
  const int h = blockIdx.x & 127;
  const size_t rowbase = (size_t)h * NCOLS + (size_t)b * LF;
  for (int l = tid; l < 4096; l += 512) {
    if (l < LF) {
      sr0[l] = Yr[rowbase + l];
      si0[l] = Yi[rowbase + l];
    } else {
      int mm = 4096 - l;
      sr0[l] = Yr[rowbase + mm];
      si0[l] = -Yi[rowbase + mm];
    }
  }
  __syncthreads();
  fft4096_lds(sr0, si0, sr1, si1, +1.0f, tid);
  float* op = out + ((size_t)(b * Hh + h)) * Ll;
  const float invL = 1.0f / 4096.0f;
  const float inv_s2 = 0.70710678118654752440f;
  for (int l = tid; l < 4096; l += 512) {
    float y = sr0[l] * invL;
    float g = 0.5f * y * (1.0f + erff(y * inv_s2));
    op[l] = g;
  }
}

extern "C" void kernel_launch(void* const* d_in, const int* in_sizes, int n_in,
                              void* d_out, int out_size, void* d_ws, size_t ws_size,
                              hipStream_t stream) {
  (void)in_sizes; (void)n_in; (void)out_size; (void)ws_size;
  const float* u      = (const float*)d_in[0];  // (B,H,L)
  const float* C_ri   = (const float*)d_in[1];  // (H,P,2)
  const float* D_ri   = (const float*)d_in[2];  // (H,H,2)
  const float* B_ri   = (const float*)d_in[3];  // (P,H,2)
  const float* Lam_ri = (const float*)d_in[4];  // (P,2)
  float* out = (float*)d_out;

  float* ws = (float*)d_ws;
  const size_t BIG = (size_t)Hh * NCOLS;  // 2,098,176 floats per plane
  float* Vr = ws;       float* Vi = Vr + BIG;
  float* Tr = Vi + BIG; float* Ti = Tr + BIG;
  float* Yr = Ti + BIG; float* Yi = Yr + BIG;
  const size_t SM = (size_t)Hh * Hh;  // 16384
  float* sm  = Yi + BIG;
  float* Wr  = sm;         float* Wi_ = Wr + SM;
  float* WIr = Wi_ + SM;   float* WIi = WIr + SM;
  float* Cr  = WIi + SM;   float* Ci  = Cr + SM;
  float* Dr  = Ci + SM;    float* Di  = Dr + SM;
  float* Br_ = Di + SM;    float* Bi_ = Br_ + SM;
  float* Bwr = Bi_ + SM;   float* Bwi = Bwr + SM;
  float* Cwr = Bwi + SM;   float* Cwi = Cwr + SM;
  float* D1r = Cwi + SM;   float* D1i = D1r + SM;
  float* Dwr = D1i + SM;   float* Dwi = Dwr + SM;

  // 1) planarize complex inputs
  const int nsm = Hh * Pp;
  deinterleave_kernel<<<(nsm + 255) / 256, 256, 0, stream>>>(C_ri, Cr, Ci, nsm);
  deinterleave_kernel<<<(nsm + 255) / 256, 256, 0, stream>>>(D_ri, Dr, Di, nsm);
  deinterleave_kernel<<<(nsm + 255) / 256, 256, 0, stream>>>(B_ri, Br_, Bi_, nsm);
  dft_build_kernel<<<(Hh * Hh + 255) / 256, 256, 0, stream>>>(Wr, Wi_, WIr, WIi);

  // 2) fold the 128-point H-axis DFT into the operators (tiny 128^3 complex GEMMs)
  dim3 gS(8, 8);
  cgemm_wmma_kernel<<<gS, 32, 0, stream>>>(Br_, Bi_, Wr, Wi_, Bwr, Bwi, 128, 0);   // Bw = B_bar * W
  cgemm_wmma_kernel<<<gS, 32, 0, stream>>>(WIr, WIi, Cr, Ci, Cwr, Cwi, 128, 0);    // Cw = W^-1 * C
  cgemm_wmma_kernel<<<gS, 32, 0, stream>>>(Dr, Di, Wr, Wi_, D1r, D1i, 128, 0);     // D1 = D * W
  cgemm_wmma_kernel<<<gS, 32, 0, stream>>>(WIr, WIi, D1r, D1i, Dwr, Dwi, 128, 0);  // Dw = W^-1 * D * W

  // 3) V = rfft_L(u)   (B-axis FFT of rfftn cancels against irfftn -> skipped)
  rfft_fwd_kernel<<<Bb * Hh, 512, 0, stream>>>(u, Vr, Vi);

  // 4) Y = Cw * (mid .* (Bw * V)) + Dw * V
  dim3 gB(8, (NCOLS + 15) / 16);
  cgemm_wmma_kernel<<<gB, 32, 0, stream>>>(Bwr, Bwi, Vr, Vi, Tr, Ti, NCOLS, 0);
  size_t tot = (size_t)Pp * NCOLS;
  mid_mul_kernel<<<(unsigned)((tot + 255) / 256), 256, 0, stream>>>(Tr, Ti, Lam_ri);
  cgemm_wmma_kernel<<<gB, 32, 0, stream>>>(Cwr, Cwi, Tr, Ti, Yr, Yi, NCOLS, 0);
  cgemm_wmma_kernel<<<gB, 32, 0, stream>>>(Dwr, Dwi, Vr, Vi, Yr, Yi, NCOLS, 1);

  // 5) y = irfft_L(Y); out = gelu(y)
  irfft_gelu_kernel<<<Bb * Hh, 512, 0, stream>>>(Yr, Yi, out);
}